// PointNetPolylineEncoder_18820546691729
// MI455X (gfx1250) — compile-verified
//
#include <hip/hip_runtime.h>
#include <hip/hip_bf16.h>

typedef __attribute__((ext_vector_type(2))) float v2f;
typedef __attribute__((ext_vector_type(8))) float v8f;

#define EPSBN 1e-5f
constexpr int Bdim = 16, Pdim = 768, Ndim = 20;
constexpr int Mrows = Bdim * Pdim * Ndim;   // 245760
static_assert(Mrows % 128 == 0, "grid must divide M");

__device__ inline v8f wmma4(v2f a, v2f b, v8f c) {
  // D = A(16x4 f32) * B(4x16 f32) + C(16x16 f32)
  return __builtin_amdgcn_wmma_f32_16x16x4_f32(
      /*neg_a=*/false, a, /*neg_b=*/false, b,
      /*c_mod=*/(short)0, c, /*reuse_a=*/false, /*reuse_b=*/false);
}

// ---------------------------------------------------------------------------
// A-fragment loader, fused per-mode transform.
// MODE 0: raw polylines [M,9] (K padded to 12 with zeros)
// MODE 1: cat(h[M,64], pooled[bp,64])   (K=128)
// MODE 2: relu(scale[k]*x + shift[k])   on bufB  (BN1 fused)
// MODE 3: mask * relu(scale[k]*x + shift[k]) on bufA (BN2 fused)
// MODE 4: relu(x) on bufB
// ---------------------------------------------------------------------------
template <int MODE>
__device__ inline v2f loadA(const float* __restrict__ Asrc,
                            const float* __restrict__ pooled,
                            const float* __restrict__ scale,
                            const float* __restrict__ shift,
                            int rowA, int bpA, float mfA, int kb) {
  v2f a;
  if constexpr (MODE == 0) {
    const float* p = Asrc + rowA * 9;
    a.x = (kb     < 9) ? p[kb]     : 0.0f;
    a.y = (kb + 1 < 9) ? p[kb + 1] : 0.0f;
  } else if constexpr (MODE == 1) {
    if (kb < 64) a = *(const v2f*)(Asrc + rowA * 64 + kb);
    else         a = *(const v2f*)(pooled + bpA * 64 + (kb - 64));
  } else if constexpr (MODE == 2 || MODE == 3) {
    v2f v = *(const v2f*)(Asrc + rowA * 64 + kb);
    a.x = fmaxf(v.x * scale[kb]     + shift[kb],     0.0f);
    a.y = fmaxf(v.y * scale[kb + 1] + shift[kb + 1], 0.0f);
    if constexpr (MODE == 3) { a.x *= mfA; a.y *= mfA; }
  } else {  // MODE 4
    v2f v = *(const v2f*)(Asrc + rowA * 64 + kb);
    a.x = fmaxf(v.x, 0.0f);
    a.y = fmaxf(v.y, 0.0f);
  }
  return a;
}

// ---------------------------------------------------------------------------
// Tiled WMMA GEMM: Out[M,ND] = transform(A)[M,KD] @ W[KREAL,ND] + bias
// 128 threads = 4 waves; each wave owns two 16-row tiles -> 128 rows / block.
// Optional epilogue: masked per-channel sum / sum-of-squares atomics (BN stats),
// MODE 4 multiplies the output row by its mask.
// ---------------------------------------------------------------------------
template <int MODE, int KD, int KREAL, int ND>
__global__ __launch_bounds__(128) void gemm_kernel(
    const float* __restrict__ Asrc, const float* __restrict__ pooled,
    const float* __restrict__ W, const float* __restrict__ bias,
    const float* __restrict__ scale, const float* __restrict__ shift,
    const unsigned char* __restrict__ mask,
    float* __restrict__ Out,
    float* __restrict__ statSum, float* __restrict__ statSq) {
  constexpr int NT = ND / 16;
  __shared__ float Blds[KD * ND];

  // Cooperative stage of W into LDS (zero-pad K rows beyond KREAL).
  for (int i = threadIdx.x; i < KD * ND; i += 128) {
    int k = i / ND;
    Blds[i] = (k < KREAL) ? W[i] : 0.0f;
  }
  __syncthreads();

  const int lane = threadIdx.x & 31;
  const int wave = threadIdx.x >> 5;
  const int lo = lane & 15;
  const int hi = lane >> 4;

  const int row0 = blockIdx.x * 128 + wave * 32;  // wave covers rows [row0, row0+32)

  // Per-lane A-row constants (A layout: M = lane%16, K = reg + 2*(lane/16)).
  const int rowA0 = row0 + lo;
  const int rowA1 = row0 + 16 + lo;
  int bp0 = 0, bp1 = 0;
  float mf0 = 1.0f, mf1 = 1.0f;
  if constexpr (MODE == 1) { bp0 = rowA0 / Ndim; bp1 = rowA1 / Ndim; }
  if constexpr (MODE == 3) {
    mf0 = mask[rowA0] ? 1.0f : 0.0f;
    mf1 = mask[rowA1] ? 1.0f : 0.0f;
  }

  v8f acc[2][NT];
#pragma unroll
  for (int mt = 0; mt < 2; ++mt)
#pragma unroll
    for (int t = 0; t < NT; ++t) acc[mt][t] = (v8f){0, 0, 0, 0, 0, 0, 0, 0};

  for (int kk = 0; kk < KD; kk += 4) {
    const int kb = kk + 2 * hi;
    v2f a0 = loadA<MODE>(Asrc, pooled, scale, shift, rowA0, bp0, mf0, kb);
    v2f a1 = loadA<MODE>(Asrc, pooled, scale, shift, rowA1, bp1, mf1, kb);
#pragma unroll
    for (int t = 0; t < NT; ++t) {
      v2f b;
      b.x = Blds[kb * ND + 16 * t + lo];
      b.y = Blds[(kb + 1) * ND + 16 * t + lo];
      acc[0][t] = wmma4(a0, b, acc[0][t]);
      acc[1][t] = wmma4(a1, b, acc[1][t]);
    }
  }

  // Epilogue: C/D layout row = reg + 8*hi, col = 16*t + lo.
#pragma unroll
  for (int mt = 0; mt < 2; ++mt) {
    const int tileRow = row0 + 16 * mt;
    const unsigned long long mm =
        *(const unsigned long long*)(mask + tileRow + 8 * hi);
#pragma unroll
    for (int t = 0; t < NT; ++t) {
      const int n = 16 * t + lo;
      const float bn = bias[n];
      float s = 0.0f, q = 0.0f;
#pragma unroll
      for (int r = 0; r < 8; ++r) {
        float v = acc[mt][t][r] + bn;
        const int row = tileRow + 8 * hi + r;
        const float mf = ((mm >> (8 * r)) & 0xffull) ? 1.0f : 0.0f;
        if constexpr (MODE == 4) v *= mf;
        Out[row * ND + n] = v;
        if constexpr (MODE <= 2) { s += mf * v; q += mf * v * v; }
      }
      if constexpr (MODE <= 2) {
        s += __shfl_xor(s, 16, 32);
        q += __shfl_xor(q, 16, 32);
        if (hi == 0) {
          unsafeAtomicAdd(&statSum[n], s);
          unsafeAtomicAdd(&statSq[n], q);
        }
      }
    }
  }
}

// Zero stats / cnt / scale-shift scratch region (first 1024 floats of ws).
__global__ __launch_bounds__(1024) void zero_kernel(float* __restrict__ ws) {
  ws[threadIdx.x] = 0.0f;
}

// Masked-count reduction: cnt = sum(mask != 0).
__global__ __launch_bounds__(256) void count_kernel(
    const unsigned char* __restrict__ mask, float* __restrict__ cnt) {
  const int i = (blockIdx.x * 256 + threadIdx.x) * 4;
  const unsigned v = *(const unsigned*)(mask + i);
  float c = (float)(((v      ) & 0xffu) != 0) + (float)(((v >> 8 ) & 0xffu) != 0) +
            (float)(((v >> 16) & 0xffu) != 0) + (float)(((v >> 24) & 0xffu) != 0);
#pragma unroll
  for (int off = 16; off > 0; off >>= 1) c += __shfl_down(c, off, 32);
  if ((threadIdx.x & 31) == 0) unsafeAtomicAdd(cnt, c);
}

// Per-channel BN finalize: scale = g*rsqrt(var+eps); shift = beta - mean*scale.
__global__ __launch_bounds__(64) void bnfin_kernel(
    const float* __restrict__ sum, const float* __restrict__ sq,
    const float* __restrict__ cnt, const float* __restrict__ g,
    const float* __restrict__ be, float* __restrict__ scale,
    float* __restrict__ shift) {
  const int c = threadIdx.x;
  const float n = fmaxf(*cnt, 1.0f);
  const float mean = sum[c] / n;
  const float var = sq[c] / n - mean * mean;
  const float s = g[c] * rsqrtf(var + EPSBN);
  scale[c] = s;
  shift[c] = be[c] - mean * s;
}

// Apply BN0+ReLU+mask in place on h, and max-pool over the N=20 points.
__global__ __launch_bounds__(64) void pool_kernel(
    float* __restrict__ bufA, const float* __restrict__ scale,
    const float* __restrict__ shift, const unsigned char* __restrict__ mask,
    float* __restrict__ pooled) {
  const int bp = blockIdx.x;     // polyline index in [0, B*P)
  const int c = threadIdx.x;     // channel
  const float s = scale[c], h = shift[c];
  float m = 0.0f;                // all h >= 0, so 0 == max over zero-padded rows
  const int base = bp * Ndim;
#pragma unroll
  for (int n = 0; n < Ndim; ++n) {
    const int row = base + n;
    float v = bufA[row * 64 + c];
    v = fmaxf(v * s + h, 0.0f);
    v *= (mask[row] ? 1.0f : 0.0f);
    bufA[row * 64 + c] = v;
    m = fmaxf(m, v);
  }
  pooled[bp * 64 + c] = m;
}

extern "C" void kernel_launch(void* const* d_in, const int* in_sizes, int n_in,
                              void* d_out, int out_size, void* d_ws, size_t ws_size,
                              hipStream_t stream) {
  const float*         poly = (const float*)d_in[0];
  const unsigned char* mask = (const unsigned char*)d_in[1];
  const float* W0 = (const float*)d_in[2];  const float* b0 = (const float*)d_in[3];
  const float* g0 = (const float*)d_in[4];  const float* be0 = (const float*)d_in[5];
  const float* W1 = (const float*)d_in[6];  const float* b1 = (const float*)d_in[7];
  const float* g1 = (const float*)d_in[8];  const float* be1 = (const float*)d_in[9];
  const float* W2 = (const float*)d_in[10]; const float* b2 = (const float*)d_in[11];
  const float* g2 = (const float*)d_in[12]; const float* be2 = (const float*)d_in[13];
  const float* W3 = (const float*)d_in[14]; const float* b3 = (const float*)d_in[15];
  const float* W4 = (const float*)d_in[16]; const float* b4 = (const float*)d_in[17];
  float* out = (float*)d_out;

  float* ws    = (float*)d_ws;
  float* stats = ws;            // sum0,sq0,sum1,sq1,sum2,sq2 : 6*64
  float* cnt   = ws + 384;
  float* sc    = ws + 400;      // scale0,shift0,scale1,shift1,scale2,shift2 : 6*64
  float* bufA  = ws + 1024;                 // [M,64]
  float* bufB  = bufA + (size_t)Mrows * 64; // [M,64]
  float* pooled = bufB + (size_t)Mrows * 64;// [B*P,64]

  const int gGemm = Mrows / 128;  // 1920

  zero_kernel<<<1, 1024, 0, stream>>>(ws);
  count_kernel<<<Mrows / (256 * 4), 256, 0, stream>>>(mask, cnt);

  // G1: x0 = polylines @ W0 + b0 ; masked stats0
  gemm_kernel<0, 12, 9, 64><<<gGemm, 128, 0, stream>>>(
      poly, nullptr, W0, b0, nullptr, nullptr, mask, bufA, stats + 0, stats + 64);
  bnfin_kernel<<<1, 64, 0, stream>>>(stats + 0, stats + 64, cnt, g0, be0, sc + 0, sc + 64);

  // BN0+ReLU+mask in place, max-pool over points
  pool_kernel<<<Bdim * Pdim, 64, 0, stream>>>(bufA, sc + 0, sc + 64, mask, pooled);

  // G2: cat(h, pooled) @ W1 + b1 ; masked stats1
  gemm_kernel<1, 128, 128, 64><<<gGemm, 128, 0, stream>>>(
      bufA, pooled, W1, b1, nullptr, nullptr, mask, bufB, stats + 128, stats + 192);
  bnfin_kernel<<<1, 64, 0, stream>>>(stats + 128, stats + 192, cnt, g1, be1, sc + 128, sc + 192);

  // G3: relu(bn1(.)) @ W2 + b2 ; masked stats2
  gemm_kernel<2, 64, 64, 64><<<gGemm, 128, 0, stream>>>(
      bufB, nullptr, W2, b2, sc + 128, sc + 192, mask, bufA, stats + 256, stats + 320);
  bnfin_kernel<<<1, 64, 0, stream>>>(stats + 256, stats + 320, cnt, g2, be2, sc + 256, sc + 320);

  // G4: (mask * relu(bn2(.))) @ W3 + b3
  gemm_kernel<3, 64, 64, 64><<<gGemm, 128, 0, stream>>>(
      bufA, nullptr, W3, b3, sc + 256, sc + 320, mask, bufB, nullptr, nullptr);

  // G5: relu(.) @ W4 + b4, row-masked, to d_out [M,128]
  gemm_kernel<4, 64, 64, 128><<<gGemm, 128, 0, stream>>>(
      bufB, nullptr, W4, b4, nullptr, nullptr, mask, out, nullptr, nullptr);
}